// M_20890720928338
// MI455X (gfx1250) — compile-verified
//
#include <hip/hip_runtime.h>
#include <hip/hip_bf16.h>
#include <math.h>

#define BATCH   512
#define CH      440
#define CRED    52
#define HW      49
#define M_TOT   (BATCH * HW)        // 25088
#define CH_HW   (CH * HW)           // 21560
#define N_TOT   (BATCH * CH * HW)   // 11038720
#define EPSV    1e-5f

#define NSLAB_COLS 112              // columns of w3 staged per block (4 slabs * 112 = 448 >= 440)
#define NT_PER_WAVE 7               // 7 N-tiles of 16 per wave
#define LDS_B_ELEMS (NSLAB_COLS * CH)   // 49280 floats = 197120 B  (needs CDNA5 320KB LDS)

typedef float v2f __attribute__((ext_vector_type(2)));
typedef float v8f __attribute__((ext_vector_type(8)));

// ---------------------------------------------------------------------------
// Kernel 1: SE gate  (one block per batch element)
//   h = relu(s @ w1^T + b1);  gate = sigmoid(h @ w2^T + b2)
//   block 0 also zeroes the global stats accumulators.
// ---------------------------------------------------------------------------
__global__ __launch_bounds__(256) void gate_kernel(
    const float* __restrict__ x178, const float* __restrict__ w1,
    const float* __restrict__ b1,   const float* __restrict__ w2,
    const float* __restrict__ b2,   float* __restrict__ gate,
    float* __restrict__ gsum,       float* __restrict__ gsq)
{
    __shared__ float s_s[CH];
    __shared__ float s_h[CRED];
    const int b   = blockIdx.x;
    const int tid = threadIdx.x;

    if (b == 0) {
        for (int t = tid; t < CH; t += 256) { gsum[t] = 0.0f; gsq[t] = 0.0f; }
    }
    for (int c = tid; c < CH; c += 256) s_s[c] = x178[b * CH + c];
    __syncthreads();

    if (tid < CRED) {
        const float* wrow = w1 + tid * CH;
        float acc = b1[tid];
        #pragma unroll 4
        for (int c = 0; c < CH; ++c) acc = fmaf(s_s[c], wrow[c], acc);
        s_h[tid] = fmaxf(acc, 0.0f);
    }
    __syncthreads();

    for (int o = tid; o < CH; o += 256) {
        const float* wrow = w2 + o * CRED;
        float acc = b2[o];
        #pragma unroll
        for (int r = 0; r < CRED; ++r) acc = fmaf(s_h[r], wrow[r], acc);
        gate[b * CH + o] = 1.0f / (1.0f + __expf(-acc));
    }
}

// ---------------------------------------------------------------------------
// Kernel 2: y = (x177 * gate) @ w3^T via V_WMMA_F32_16X16X4_F32.
// grid.x = 4 N-slabs (112 cols each), grid.y = 196 M-blocks.
// Block = 256 threads = 8 waves. Block stages its w3 slab (112 x 440 = 197KB)
// into LDS once; each wave computes one 16(M) x 112(N) strip: 7 accumulators,
// A-fragment loaded once per K-step and reused across 7 WMMAs, B-fragments
// read from LDS as ds_load_b64 ([col][k] layout -> K-components contiguous).
// ---------------------------------------------------------------------------
__global__ __launch_bounds__(256) void gemm_kernel(
    const float* __restrict__ x177, const float* __restrict__ gate,
    const float* __restrict__ w3,   float* __restrict__ y,
    float* __restrict__ gsum,       float* __restrict__ gsq)
{
    extern __shared__ float smem[];
    float* lds_b = smem;                    // [col][k] : col*CH + k
    float* lsum  = smem + LDS_B_ELEMS;      // [112]
    float* lsq   = lsum + NSLAB_COLS;       // [112]

    const int tid  = threadIdx.x;
    const int lane = tid & 31;
    const int wave = tid >> 5;
    const int l15  = lane & 15;
    const int lhi  = lane >> 4;             // half-wave: 0 or 1
    const int kp   = lhi * 2;               // this lane's K-slot pair within a K=4 step

    const int slab_base = blockIdx.x * NSLAB_COLS;
    const int m_base    = (blockIdx.y * 8 + wave) * 16;

    // ---- zero block-local stats
    for (int t = tid; t < NSLAB_COLS; t += 256) { lsum[t] = 0.0f; lsq[t] = 0.0f; }

    // ---- stage w3 slab into LDS (rows slab_base..slab_base+111, clamped; masked later)
    for (int i = tid; i < LDS_B_ELEMS; i += 256) {
        const int col = i / CH;
        const int k   = i - col * CH;
        const int o   = slab_base + col;
        const int oc  = (o < CH) ? o : (CH - 1);
        lds_b[i] = w3[(size_t)oc * CH + k];
    }
    __syncthreads();

    // ---- A operand addressing: row m = m_base + l15
    const unsigned mrow = (unsigned)(m_base + l15);
    const unsigned bidx = mrow / HW;
    const unsigned hw   = mrow - bidx * HW;
    const float* arow = x177 + (size_t)bidx * CH_HW + hw;   // element c at arow[c*HW]
    const float* grow = gate + (size_t)bidx * CH;

    const float* bbase = lds_b + l15 * CH + kp;             // + j*16*CH + kc

    v8f acc[NT_PER_WAVE];
    #pragma unroll
    for (int j = 0; j < NT_PER_WAVE; ++j) acc[j] = (v8f){};

    #pragma unroll 2
    for (int kc = 0; kc < CH; kc += 4) {
        const int c0 = kc + kp;
        v2f af;
        af.x = arow[(size_t)c0 * HW]       * grow[c0];
        af.y = arow[(size_t)(c0 + 1) * HW] * grow[c0 + 1];
        const float* bp = bbase + kc;
        #pragma unroll
        for (int j = 0; j < NT_PER_WAVE; ++j) {
            const v2f bf = *(const v2f*)(bp + j * (16 * CH));
            acc[j] = __builtin_amdgcn_wmma_f32_16x16x4_f32(
                false, af, false, bf, (short)0, acc[j], false, false);
        }
    }

    // ---- epilogue: D layout -> lane holds col (base + l15), rows m_base + 8*lhi + r
    const int mstart = m_base + 8 * lhi;
    size_t offr[8];
    #pragma unroll
    for (int r = 0; r < 8; ++r) {
        const unsigned m  = (unsigned)(mstart + r);
        const unsigned bb = m / HW;
        const unsigned hh = m - bb * HW;
        offr[r] = (size_t)bb * CH_HW + hh;
    }

    #pragma unroll
    for (int j = 0; j < NT_PER_WAVE; ++j) {
        const int ocol = slab_base + j * 16 + l15;
        if (ocol < CH) {
            float s = 0.0f, q = 0.0f;
            #pragma unroll
            for (int r = 0; r < 8; ++r) {
                const float v = acc[j][r];
                s += v;
                q = fmaf(v, v, q);
                y[offr[r] + (size_t)ocol * HW] = v;
            }
            atomicAdd(&lsum[j * 16 + l15], s);
            atomicAdd(&lsq[j * 16 + l15], q);
        }
    }
    __syncthreads();

    for (int t = tid; t < NSLAB_COLS; t += 256) {
        const int oc = slab_base + t;
        if (oc < CH) {
            atomicAdd(&gsum[oc], lsum[t]);
            atomicAdd(&gsq[oc],  lsq[t]);
        }
    }
}

// ---------------------------------------------------------------------------
// Kernel 3: finalize per-channel scale/shift
// ---------------------------------------------------------------------------
__global__ __launch_bounds__(512) void stats_kernel(
    const float* __restrict__ gsum, const float* __restrict__ gsq,
    const float* __restrict__ gamma, const float* __restrict__ beta,
    float* __restrict__ scale, float* __restrict__ shift)
{
    const int t = threadIdx.x;
    if (t < CH) {
        const float inv_n = 1.0f / (float)M_TOT;
        const float mean  = gsum[t] * inv_n;
        const float var   = gsq[t] * inv_n - mean * mean;
        const float inv   = rsqrtf(var + EPSV);
        const float sc    = inv * gamma[t];
        scale[t] = sc;
        shift[t] = beta[t] - mean * sc;
    }
}

// ---------------------------------------------------------------------------
// Kernel 4: elementwise normalize, float4 vectorized (N_TOT % 1024 == 0)
// ---------------------------------------------------------------------------
__global__ __launch_bounds__(256) void norm_kernel(
    const float* __restrict__ y, const float* __restrict__ scale,
    const float* __restrict__ shift, float* __restrict__ out)
{
    const size_t i0 = ((size_t)blockIdx.x * 256 + threadIdx.x) * 4;
    const float4 v = *(const float4*)(y + i0);
    float4 r;
    {
        const unsigned o0 = (unsigned)((i0 / HW) % CH);
        r.x = fmaf(v.x, scale[o0], shift[o0]);
    }
    {
        const unsigned o1 = (unsigned)(((i0 + 1) / HW) % CH);
        r.y = fmaf(v.y, scale[o1], shift[o1]);
    }
    {
        const unsigned o2 = (unsigned)(((i0 + 2) / HW) % CH);
        r.z = fmaf(v.z, scale[o2], shift[o2]);
    }
    {
        const unsigned o3 = (unsigned)(((i0 + 3) / HW) % CH);
        r.w = fmaf(v.w, scale[o3], shift[o3]);
    }
    *(float4*)(out + i0) = r;
}

// ---------------------------------------------------------------------------
extern "C" void kernel_launch(void* const* d_in, const int* in_sizes, int n_in,
                              void* d_out, int out_size, void* d_ws, size_t ws_size,
                              hipStream_t stream)
{
    const float* x178  = (const float*)d_in[0];
    const float* x177  = (const float*)d_in[1];
    const float* w1    = (const float*)d_in[2];
    const float* b1    = (const float*)d_in[3];
    const float* w2    = (const float*)d_in[4];
    const float* b2    = (const float*)d_in[5];
    const float* w3    = (const float*)d_in[6];
    const float* gamma = (const float*)d_in[7];
    const float* beta  = (const float*)d_in[8];
    float* out = (float*)d_out;

    float* ws    = (float*)d_ws;
    float* gate  = ws;                        // 512*440
    float* ybuf  = gate + (size_t)BATCH * CH; // 512*440*49
    float* gsum  = ybuf + (size_t)N_TOT;      // 440
    float* gsq   = gsum + CH;                 // 440
    float* scale = gsq + CH;                  // 440
    float* shift = scale + CH;                // 440

    gate_kernel<<<BATCH, 256, 0, stream>>>(x178, w1, b1, w2, b2, gate, gsum, gsq);

    const size_t lds_bytes = (LDS_B_ELEMS + 2 * NSLAB_COLS) * sizeof(float); // 198016 B
    dim3 ggrid(4, M_TOT / (16 * 8));          // 4 N-slabs x 196 M-blocks
    gemm_kernel<<<ggrid, 256, lds_bytes, stream>>>(x177, gate, w3, ybuf, gsum, gsq);

    stats_kernel<<<1, 512, 0, stream>>>(gsum, gsq, gamma, beta, scale, shift);

    norm_kernel<<<N_TOT / 1024, 256, 0, stream>>>(ybuf, scale, shift, out);
}